// PositionId_80668075753523
// MI455X (gfx1250) — compile-verified
//
#include <hip/hip_runtime.h>

// 128-bit vector types.
typedef float v4f __attribute__((ext_vector_type(4)));
typedef int   v4i __attribute__((vector_size(16)));   // matches async-builtin param type

#define AS1 __attribute__((address_space(1)))   // global
#define AS3 __attribute__((address_space(3)))   // LDS

#if __has_builtin(__builtin_amdgcn_global_load_async_to_lds_b128) && \
    __has_builtin(__builtin_amdgcn_global_store_async_from_lds_b128)
#define USE_ASYNC_LDS 1
#else
#define USE_ASYNC_LDS 0
#endif

static __device__ __forceinline__ void wait_asynccnt0() {
#if __has_builtin(__builtin_amdgcn_s_wait_asynccnt)
    __builtin_amdgcn_s_wait_asynccnt(0);
#else
    asm volatile("s_wait_asynccnt 0" ::: "memory");
#endif
}

// ---------------------------------------------------------------------------
// Specialized kernel: N_SEG = 16 segments (reference BATCH), D_MODEL = 1024.
// Each block moves ROWS rows of 4 KB. 256 threads x 16 B = 4 KB per row.
// Per wave: 4 async loads in flight -> one s_wait_asynccnt 0 -> 4 NT stores.
// Segment search is uniform in blockIdx -> branch-free s_cmp/s_cselect chain.
// ---------------------------------------------------------------------------
constexpr int ROWS = 4;

__global__ void __launch_bounds__(256)
ragged_gather_spec(const float* __restrict__ buf,
                   const int*   __restrict__ offs,
                   float*       __restrict__ out,
                   int total_rows) {
#if USE_ASYNC_LDS
    __shared__ __align__(16) char smem[ROWS * 4096];
#endif
    const int t    = threadIdx.x;
    const int row0 = blockIdx.x * ROWS;

    // Bulk-load the 16 upper offsets once (uniform -> scalar loads).
    int o[16];
#pragma unroll
    for (int i = 0; i < 16; ++i) o[i] = offs[i + 1];

    int pos[ROWS];
#pragma unroll
    for (int j = 0; j < ROWS; ++j) {
        const int row = row0 + j;
        int base = 0;
#pragma unroll
        for (int i = 0; i < 16; ++i) base = (row >= o[i]) ? o[i] : base;
        pos[j] = row - base;                      // local position in segment
    }

#if USE_ASYNC_LDS
#pragma unroll
    for (int j = 0; j < ROWS; ++j) {
        if (row0 + j < total_rows) {
            const float* src = buf + (size_t)pos[j] * 1024u;
            AS1 v4i* g = (AS1 v4i*)(src + (size_t)t * 4);
            AS3 v4i* l = (AS3 v4i*)(smem + j * 4096 + t * 16);
            __builtin_amdgcn_global_load_async_to_lds_b128(g, l, 0, 0);
        }
    }
    wait_asynccnt0();
#pragma unroll
    for (int j = 0; j < ROWS; ++j) {
        const int row = row0 + j;
        if (row < total_rows) {
            AS1 v4i* g = (AS1 v4i*)(out + (size_t)row * 1024u + (size_t)t * 4);
            AS3 v4i* l = (AS3 v4i*)(smem + j * 4096 + t * 16);
            // cpol=1 -> TH=NT: stream 256MB of output past L2 so it doesn't
            // evict the hot 32MB position buffer (which is read ~8x).
            __builtin_amdgcn_global_store_async_from_lds_b128(g, l, 0, 1);
        }
    }
    // s_endpgm's implicit wait-idle drains ASYNCcnt before the wave exits.
#else
#pragma unroll
    for (int j = 0; j < ROWS; ++j) {
        const int row = row0 + j;
        if (row < total_rows) {
            const v4f v = ((const v4f*)(buf + (size_t)pos[j] * 1024u))[t];
            __builtin_nontemporal_store(v, (v4f*)(out + (size_t)row * 1024u) + t);
        }
    }
#endif
}

// ---------------------------------------------------------------------------
// Generic fallback for any segment count (one row per block).
// ---------------------------------------------------------------------------
__global__ void __launch_bounds__(256)
ragged_gather_generic(const float* __restrict__ buf,
                      const int*   __restrict__ offs,
                      float*       __restrict__ out,
                      int n_off) {
    const int row = blockIdx.x;
    int base = 0;
    for (int i = 1; i < n_off; ++i) {
        const int o = offs[i];
        base = (row >= o) ? o : base;
    }
    const float* src = buf + (size_t)(row - base) * 1024u;
    float*       dst = out + (size_t)row * 1024u;
    const int t = threadIdx.x;

#if USE_ASYNC_LDS
    __shared__ __align__(16) char smem[4096];
    AS1 v4i* gsrc = (AS1 v4i*)(src + (size_t)t * 4);
    AS1 v4i* gdst = (AS1 v4i*)(dst + (size_t)t * 4);
    AS3 v4i* lds  = (AS3 v4i*)(smem + t * 16);
    __builtin_amdgcn_global_load_async_to_lds_b128(gsrc, lds, 0, 0);
    wait_asynccnt0();
    __builtin_amdgcn_global_store_async_from_lds_b128(gdst, lds, 0, 1);
#else
    const v4f v = ((const v4f*)src)[t];
    __builtin_nontemporal_store(v, ((v4f*)dst) + t);
#endif
}

extern "C" void kernel_launch(void* const* d_in, const int* in_sizes, int n_in,
                              void* d_out, int out_size, void* d_ws, size_t ws_size,
                              hipStream_t stream) {
    const float* buf  = (const float*)d_in[0];   // [8192, 1024] f32
    const int*   offs = (const int*)d_in[1];     // [17] i32 ragged offsets
    float*       out  = (float*)d_out;           // [65536, 1024] f32

    const int n_off   = in_sizes[1];             // 17
    const int d_model = 1024;                    // D_MODEL (fixed by reference)
    const int rows    = out_size / d_model;      // 65536 total tokens

    if (n_off == 17) {
        const int blocks = (rows + ROWS - 1) / ROWS;   // 16384
        ragged_gather_spec<<<blocks, 256, 0, stream>>>(buf, offs, out, rows);
    } else {
        ragged_gather_generic<<<rows, 256, 0, stream>>>(buf, offs, out, n_off);
    }
}